// MyModel_80410377716427
// MI455X (gfx1250) — compile-verified
//
#include <hip/hip_runtime.h>
#include <stdint.h>

#define BATCH     1000000
#define NV        9
#define F_IN      5
#define F_HID     3
#define TILE      128                         // samples per block (4 waves)
#define A_PER_S   (NV * NV)                   // 81 floats
#define H_PER_S   (NV * F_IN)                 // 45 floats
#define A4_TILE   (TILE * A_PER_S / 4)        // 2592 float4 per tile
#define H4_TILE   (TILE * H_PER_S / 4)        // 1440 float4 per tile

#if defined(__HIP_DEVICE_COMPILE__) && __has_builtin(__builtin_amdgcn_global_load_async_to_lds_b128)
#define HAVE_ASYNC_LDS 1
#else
#define HAVE_ASYNC_LDS 0
#endif

#if defined(__HIP_DEVICE_COMPILE__) && __has_builtin(__builtin_amdgcn_wmma_f32_16x16x4_f32)
#define HAVE_WMMA_F32 1
#else
#define HAVE_WMMA_F32 0
#endif

typedef float v2f __attribute__((ext_vector_type(2)));
typedef float v8f __attribute__((ext_vector_type(8)));

#if HAVE_ASYNC_LDS
typedef int v4i_vs __attribute__((vector_size(4 * sizeof(int))));
typedef __attribute__((address_space(1))) v4i_vs* g_v4i_ptr;   // global (__device__) AS
typedef __attribute__((address_space(3))) v4i_vs* l_v4i_ptr;   // LDS (__shared__) AS
#endif

__device__ __forceinline__ void async_copy16(const void* g, void* l) {
#if HAVE_ASYNC_LDS
    // global_load_async_to_lds_b128: copies 16 bytes from the global address g to the LDS address l; tracked by ASYNCcnt
    __builtin_amdgcn_global_load_async_to_lds_b128(
        (g_v4i_ptr)(uintptr_t)g,
        (l_v4i_ptr)(uint32_t)(uintptr_t)l,
        /*imm offset*/0, /*cpol*/0);
#else
    *(float4*)l = *(const float4*)g;
#endif
}

__device__ __forceinline__ void async_wait0() {
#if HAVE_ASYNC_LDS
#if __has_builtin(__builtin_amdgcn_s_wait_asynccnt)
    __builtin_amdgcn_s_wait_asynccnt(0);
#else
    asm volatile("s_wait_asynccnt 0" ::: "memory");
#endif
#endif
}

__global__ __launch_bounds__(TILE) void gcn_head_kernel(
    const float* __restrict__ A,     // [B,9,9]
    const float* __restrict__ H0,    // [B,9,5]
    const float* __restrict__ W,     // [5,3]
    const float* __restrict__ FCW,   // [3,1]
    const float* __restrict__ FCB,   // [1]
    float* __restrict__ out)         // [B,1]
{
    __shared__ float4 ldsA[A4_TILE];   // 41,472 B
    __shared__ float4 ldsH[H4_TILE];   // 23,040 B

    const int tid  = threadIdx.x;
    const int tile = blockIdx.x;

    // ---- cooperative, coalesced async global->LDS staging (b128) ----
    {
        const size_t baseA4 = (size_t)tile * A4_TILE;
        const size_t totA4  = (size_t)BATCH * A_PER_S / 4;   // 20,250,000
        const float4* gA = (const float4*)A;
        for (int i = tid; i < A4_TILE; i += TILE)
            if (baseA4 + (size_t)i < totA4)
                async_copy16(gA + baseA4 + i, &ldsA[i]);

        const size_t baseH4 = (size_t)tile * H4_TILE;
        const size_t totH4  = (size_t)BATCH * H_PER_S / 4;   // 11,250,000
        const float4* gH = (const float4*)H0;
        for (int i = tid; i < H4_TILE; i += TILE)
            if (baseH4 + (size_t)i < totH4)
                async_copy16(gH + baseH4 + i, &ldsH[i]);
    }
    async_wait0();
    __syncthreads();

    // NOTE: no early return — EXEC must stay all-1s for the WMMA head.
    const size_t b = (size_t)tile * TILE + tid;

    // ---- uniform params (L1-resident) ----
    float Wm[F_IN][F_HID];
#pragma unroll
    for (int f = 0; f < F_IN; ++f)
#pragma unroll
        for (int h = 0; h < F_HID; ++h)
            Wm[f][h] = W[f * F_HID + h];
    const float fw0 = FCW[0], fw1 = FCW[1], fw2 = FCW[2];
    const float fb = FCB[0];

    const float* a  = (const float*)ldsA + (size_t)tid * A_PER_S;  // lane-stride 81 ≡ 17 mod 64 banks
    const float* h0 = (const float*)ldsH + (size_t)tid * H_PER_S;

    // ---- A -> registers, fused column-degree accumulation (deg = 1 + colsum(A)) ----
    float av[A_PER_S];
    float deg[NV];
#pragma unroll
    for (int j = 0; j < NV; ++j) deg[j] = 1.0f;   // +I contribution
#pragma unroll
    for (int i = 0; i < NV; ++i)
#pragma unroll
        for (int j = 0; j < NV; ++j) {
            const float v = a[i * NV + j];
            av[i * NV + j] = v;
            deg[j] += v;
        }

    float dinv[NV];
#pragma unroll
    for (int j = 0; j < NV; ++j)
        dinv[j] = 1.0f / sqrtf(deg[j]);

    // ---- g = diag(dinv) * (h0 @ W) ----
    float g0[NV], g1[NV], g2[NV];
#pragma unroll
    for (int j = 0; j < NV; ++j) {
        float hw0 = 0.f, hw1 = 0.f, hw2 = 0.f;
#pragma unroll
        for (int f = 0; f < F_IN; ++f) {
            const float x = h0[j * F_IN + f];
            hw0 = fmaf(x, Wm[f][0], hw0);
            hw1 = fmaf(x, Wm[f][1], hw1);
            hw2 = fmaf(x, Wm[f][2], hw2);
        }
        const float s = dinv[j];
        g0[j] = s * hw0; g1[j] = s * hw1; g2[j] = s * hw2;
    }

    // ---- rows: h2 = sum_i relu(dinv[i] * (A[i,:]·g + g[i,:])) ----
    float h20 = 0.f, h21 = 0.f, h22 = 0.f;
#pragma unroll
    for (int i = 0; i < NV; ++i) {
        float t0 = g0[i], t1 = g1[i], t2 = g2[i];   // identity (A+I) term
#pragma unroll
        for (int j = 0; j < NV; ++j) {
            const float w = av[i * NV + j];
            t0 = fmaf(w, g0[j], t0);
            t1 = fmaf(w, g1[j], t1);
            t2 = fmaf(w, g2[j], t2);
        }
        const float s = dinv[i];
        t0 *= s; t1 *= s; t2 *= s;
        h20 += t0 > 0.f ? t0 : 0.f;
        h21 += t1 > 0.f ? t1 : 0.f;
        h22 += t2 > 0.f ? t2 : 0.f;
    }

#if HAVE_WMMA_F32
    // ---- regression head as fp32 WMMA: D(16x16) = h2(16x4) x fcw(4x16) + fc_b ----
    // Two WMMAs per wave cover its 32 samples (sample index within wave == lane).
    const int lane = tid & 31;
    const bool lo = (lane < 16);

    v8f c;
#pragma unroll
    for (int k = 0; k < 8; ++k) c[k] = fb;

    // B(4x16): vgpr0 = K0 (lanes 0-15) / K2 (lanes 16-31); vgpr1 = K1 / K3(pad)
    v2f bm;
    bm[0] = lo ? fw0 : fw2;
    bm[1] = lo ? fw1 : 0.f;

    // A(16x4) for samples 0..15 of the wave
    const int srcA = lane & 15;
    const float x0 = __shfl(h20, srcA, 32);
    const float x1 = __shfl(h21, srcA, 32);
    const float x2 = __shfl(h22, srcA, 32);
    v2f a1; a1[0] = lo ? x0 : x2; a1[1] = lo ? x1 : 0.f;

    // A(16x4) for samples 16..31 of the wave
    const int srcB = 16 | (lane & 15);
    const float y0 = __shfl(h20, srcB, 32);
    const float y1 = __shfl(h21, srcB, 32);
    const float y2 = __shfl(h22, srcB, 32);
    v2f a2; a2[0] = lo ? y0 : y2; a2[1] = lo ? y1 : 0.f;

    const v8f d1 = __builtin_amdgcn_wmma_f32_16x16x4_f32(
        false, a1, false, bm, (short)0, c, false, false);
    const v8f d2 = __builtin_amdgcn_wmma_f32_16x16x4_f32(
        false, a2, false, bm, (short)0, c, false, false);

    // Extract own sample: lane m holds D[(m&7)+8*(m/16)][m%16] in d[m&7].
    const int r = lane & 7;
    float e1 = d1[0], e2 = d2[0];
#pragma unroll
    for (int k = 1; k < 8; ++k) {
        e1 = (r == k) ? d1[k] : e1;
        e2 = (r == k) ? d2[k] : e2;
    }
    const int srcD = (lane & 7) | ((lane & 8) << 1);
    const float v1 = __shfl(e1, srcD, 32);
    const float v2 = __shfl(e2, srcD, 32);
    const float result = lo ? v1 : v2;
#else
    float result = fb;
    result = fmaf(h20, fw0, result);
    result = fmaf(h21, fw1, result);
    result = fmaf(h22, fw2, result);
#endif

    if (b < BATCH) out[b] = result;   // coalesced b32 store
}

extern "C" void kernel_launch(void* const* d_in, const int* in_sizes, int n_in,
                              void* d_out, int out_size, void* d_ws, size_t ws_size,
                              hipStream_t stream) {
    const float* A   = (const float*)d_in[0];
    const float* H0  = (const float*)d_in[1];
    const float* W   = (const float*)d_in[2];
    const float* FCW = (const float*)d_in[3];
    const float* FCB = (const float*)d_in[4];
    float* out = (float*)d_out;

    const int grid = (BATCH + TILE - 1) / TILE;   // 7813 blocks
    hipLaunchKernelGGL(gcn_head_kernel, dim3(grid), dim3(TILE), 0, stream,
                       A, H0, W, FCW, FCB, out);
}